// Loss_9749575762182
// MI455X (gfx1250) — compile-verified
//
#include <hip/hip_runtime.h>

// ext vector types (native clang vectors -> map to consecutive VGPRs)
typedef float f4  __attribute__((ext_vector_type(4)));
typedef float v2f __attribute__((ext_vector_type(2)));
typedef float v8f __attribute__((ext_vector_type(8)));

#define EPS_F 1e-10f
#define THREADS 256
#define BLOCKS 2048

__device__ __forceinline__ float bce_abs(float t, float p, float o) {
    // t*log(p+eps) + (1-t)*log(1-p+eps), then |.*o|
    float b = t * __logf(p + EPS_F) + (1.0f - t) * __logf(1.0f - p + EPS_F);
    return fabsf(b * o);
}

// Pass 1: grid-stride streaming reduction. 128-bit NT loads on all 5 streams,
// per-thread fp32 accumulation, LDS fold 256->32, then a WMMA f32 16x16x4
// cross-lane reduction producing (wmse, wcl) block partials.
__global__ __launch_bounds__(THREADS) void loss_partial_kernel(
    const float* __restrict__ input, const float* __restrict__ target,
    const float* __restrict__ weight, const float* __restrict__ sub_target,
    const float* __restrict__ target_pre, const float* __restrict__ sub_obrT,
    float* __restrict__ partials, int n4, int n_total, int Dcols)
{
    const int tid    = blockIdx.x * blockDim.x + threadIdx.x;
    const int stride = gridDim.x * blockDim.x;
    const int D4     = Dcols >> 2;   // D is a multiple of 4 (2048)

    const f4* __restrict__ in4 = (const f4*)input;
    const f4* __restrict__ tg4 = (const f4*)target;
    const f4* __restrict__ st4 = (const f4*)sub_target;
    const f4* __restrict__ tp4 = (const f4*)target_pre;
    const f4* __restrict__ ob4 = (const f4*)sub_obrT;

    float acc0 = 0.0f;   // wmse partial
    float acc1 = 0.0f;   // wcl partial

    for (int i = tid; i < n4; i += stride) {
        f4 a = __builtin_nontemporal_load(in4 + i);   // global_load_b128 NT
        f4 b = __builtin_nontemporal_load(tg4 + i);
        f4 t = __builtin_nontemporal_load(st4 + i);
        f4 p = __builtin_nontemporal_load(tp4 + i);
        f4 o = __builtin_nontemporal_load(ob4 + i);
        float w = weight[i / D4];                      // 32KB vector, cache-hot
        f4 d = (b - a) * w;
        acc0 += d.x * d.x + d.y * d.y + d.z * d.z + d.w * d.w;
        acc1 += bce_abs(t.x, p.x, o.x) + bce_abs(t.y, p.y, o.y)
              + bce_abs(t.z, p.z, o.z) + bce_abs(t.w, p.w, o.w);
    }

    // scalar tail (n_total % 4), handled once
    if (tid == 0) {
        for (int e = n4 * 4; e < n_total; ++e) {
            float d = (target[e] - input[e]) * weight[e / Dcols];
            acc0 += d * d;
            acc1 += bce_abs(sub_target[e], target_pre[e], sub_obrT[e]);
        }
    }

    __shared__ float s0[THREADS];
    __shared__ float s1[THREADS];
    s0[threadIdx.x] = acc0;
    s1[threadIdx.x] = acc1;
    __syncthreads();

    if (threadIdx.x < 32) {            // whole wave0 -> EXEC all-1s for WMMA
        const int lane = threadIdx.x;
        float p0 = 0.0f, p1 = 0.0f;
        #pragma unroll
        for (int k = 0; k < THREADS / 32; ++k) {
            p0 += s0[lane + 32 * k];
            p1 += s1[lane + 32 * k];
        }

        // A (16x4 f32): VGPR0 = wmse partial, VGPR1 = wcl partial.
        //   lanes 0-15  -> A[m][0]=p0, A[m][1]=p1
        //   lanes 16-31 -> A[m][2]=p0, A[m][3]=p1
        v2f A; A.x = p0; A.y = p1;
        // B (4x16 f32): VGPR0: lanes0-15 K=0 / lanes16-31 K=2,
        //               VGPR1: lanes0-15 K=1 / lanes16-31 K=3.
        // Column 0 sums wmse partials (B[0][0]=B[2][0]=1),
        // column 1 sums wcl partials  (B[1][1]=B[3][1]=1).
        v2f B;
        B.x = (lane == 0 || lane == 16) ? 1.0f : 0.0f;
        B.y = (lane == 1 || lane == 17) ? 1.0f : 0.0f;
        v8f C = {};
        // v_wmma_f32_16x16x4_f32: D[m][0] = sum of wmse over K, D[m][1] = wcl
        v8f Dm = __builtin_amdgcn_wmma_f32_16x16x4_f32(
            /*neg_a=*/false, A, /*neg_b=*/false, B,
            /*c_mod=*/(short)0, C, /*reuse_a=*/false, /*reuse_b=*/false);

        // lane0 holds D[0..7][0], lane16 holds D[8..15][0] (wmse);
        // lane1/lane17 same for column 1 (wcl).
        float s = Dm[0] + Dm[1] + Dm[2] + Dm[3] + Dm[4] + Dm[5] + Dm[6] + Dm[7];
        s += __shfl_xor(s, 16, 32);
        if (lane == 0) partials[2 * blockIdx.x + 0] = s;
        if (lane == 1) partials[2 * blockIdx.x + 1] = s;
    }
}

// Pass 2: deterministic final reduction of BLOCKS partial pairs.
__global__ __launch_bounds__(THREADS) void loss_final_kernel(
    const float* __restrict__ partials, float* __restrict__ out,
    int nblocks, float inv_count)
{
    __shared__ float s0[THREADS];
    __shared__ float s1[THREADS];
    float a = 0.0f, b = 0.0f;
    for (int i = threadIdx.x; i < nblocks; i += THREADS) {
        a += partials[2 * i + 0];
        b += partials[2 * i + 1];
    }
    s0[threadIdx.x] = a;
    s1[threadIdx.x] = b;
    __syncthreads();
    for (int s = THREADS / 2; s > 0; s >>= 1) {
        if (threadIdx.x < s) {
            s0[threadIdx.x] += s0[threadIdx.x + s];
            s1[threadIdx.x] += s1[threadIdx.x + s];
        }
        __syncthreads();
    }
    if (threadIdx.x == 0) {
        out[0] = s0[0] * inv_count;   // wmse
        out[1] = s1[0] * inv_count;   // wcl
    }
}

extern "C" void kernel_launch(void* const* d_in, const int* in_sizes, int n_in,
                              void* d_out, int out_size, void* d_ws, size_t ws_size,
                              hipStream_t stream) {
    const float* input      = (const float*)d_in[0];
    const float* target     = (const float*)d_in[1];
    const float* weight     = (const float*)d_in[2];
    const float* sub_target = (const float*)d_in[3];
    const float* target_pre = (const float*)d_in[4];
    const float* sub_obrT   = (const float*)d_in[5];

    const int n_total = in_sizes[0];          // N*D
    const int Nrows   = in_sizes[2];          // N
    const int Dcols   = n_total / Nrows;      // D
    const int n4      = n_total / 4;

    float* partials = (float*)d_ws;           // 2*BLOCKS floats (16 KB)

    loss_partial_kernel<<<BLOCKS, THREADS, 0, stream>>>(
        input, target, weight, sub_target, target_pre, sub_obrT,
        partials, n4, n_total, Dcols);

    loss_final_kernel<<<1, THREADS, 0, stream>>>(
        partials, (float*)d_out, BLOCKS, 1.0f / (float)n_total);
}